// RoIAlignAvg_43241730736873
// MI455X (gfx1250) — compile-verified
//
#include <hip/hip_runtime.h>

// RoI-Align-Avg as per-RoI two-sided GEMM on 16x16 feature tiles, executed with
// V_WMMA_F32_16X16X4_F32 (f32 in / f32 acc, K=16 as 4 x K=4 chain).
//
//   out[c, oi, oj] = sum_{th,tw} Uh[oi,th] * F[c, h0+th, w0+tw] * Uw[oj,tw]
//
// Uh/Uw (7x16, zero padded to 16x16) fold in bilinear weights, the validity
// mask (factorizes per-axis), and the final 2x2 average (0.5*(row+row')).
//
// v5: removed the output LDS staging entirely. The stage-2 D layout already
// holds out[g][n] in VGPR g / lane n, so lanes 0..6 store rows directly with
// saddr + scale_offset + constant-immediate addresses. This kills the in-loop
// div-by-7 lowering, 10 DS ops, and 2 fences per channel. Steady state per
// channel: 8 pipelined loads, 4 WMMA, 4 ds_store_2addr, fence, 4 ds_load_2addr,
// 4 WMMA, 7 immediate-offset stores.

typedef __attribute__((ext_vector_type(2))) float v2f;
typedef __attribute__((ext_vector_type(8))) float v8f;

#define SCALEF 0.0625f
#define Cc 512
#define Hc 100
#define Wc 100
#define HW (Hc * Wc)
#define TILE 16
#define PITCH 17          // +1 pad: conflict-free LDS columns
#define NWAVES 8
#define CH_PER_BLOCK 64
#define CH_PER_WAVE (CH_PER_BLOCK / NWAVES)

// Zero-instruction compiler code-motion fence. Cross-lane LDS RAW within one
// wave is already ordered by hardware (DS ops per-wave are in order); we only
// need to stop the compiler from reordering DS ops around the relayout point.
__device__ __forceinline__ void lds_order_fence() {
  __builtin_amdgcn_wave_barrier();
}

__device__ __forceinline__ int imin(int a, int b) { return a < b ? a : b; }

__global__ __launch_bounds__(256) void RoIAlignAvg_wmma_kernel(
    const float* __restrict__ feat, const float* __restrict__ rois,
    float* __restrict__ out) {
  __shared__ float Uh[TILE][PITCH];
  __shared__ float Uw[TILE][PITCH];
  __shared__ float Wh[8][PITCH];
  __shared__ float Ww[8][PITCH];
  __shared__ float M1[NWAVES][TILE][PITCH];
  __shared__ int sh0, sw0, sb;

  const int r = blockIdx.y;
  const int cbase = blockIdx.x * CH_PER_BLOCK;
  const int tid = threadIdx.x;

  // ---- zero the weight scratch ----
  {
    float* p0 = &Uh[0][0];
    for (int i = tid; i < TILE * PITCH; i += 256) p0[i] = 0.f;
    float* p1 = &Uw[0][0];
    for (int i = tid; i < TILE * PITCH; i += 256) p1[i] = 0.f;
    float* p2 = &Wh[0][0];
    for (int i = tid; i < 8 * PITCH; i += 256) p2[i] = 0.f;
    float* p3 = &Ww[0][0];
    for (int i = tid; i < 8 * PITCH; i += 256) p3[i] = 0.f;
  }
  __syncthreads();

  // ---- per-RoI weight construction (threads 0..15) ----
  if (tid < 16) {
    const float* rp = rois + (size_t)r * 5;
    float xs = rp[1] * SCALEF, ys = rp[2] * SCALEF;
    float xe = rp[3] * SCALEF, ye = rp[4] * SCALEF;
    float rwd = fmaxf(xe - xs + 1.f, 0.f);
    float rht = fmaxf(ye - ys + 1.f, 0.f);
    const bool isH = tid < 8;
    const int i = tid & 7;
    float start = isH ? ys : xs;
    float binsz = (isH ? rht : rwd) / 7.f;   // ah-1 == aw-1 == 7
    float lim = isH ? (float)Hc : (float)Wc;
    float v = start + (float)i * binsz;
    float valid = (v >= 0.f && v < lim) ? 1.f : 0.f;
    float fs = fminf(fmaxf(floorf(v), 0.f), lim - 2.f);
    float fr = v - fs;                        // can exceed 1 when clipped (matches ref)
    int idx = (int)fs;
    // tile origin = clipped floor of the first sample (monotone => minimum)
    int base0 = (int)fminf(fmaxf(floorf(start), 0.f), lim - 2.f);
    int t = idx - base0;                      // 0..14 by span bound (<=13.5+1)
    if (isH) {
      Wh[i][t]     += valid * (1.f - fr);
      Wh[i][t + 1] += valid * fr;
      if (i == 0) sh0 = base0;
    } else {
      Ww[i][t]     += valid * (1.f - fr);
      Ww[i][t + 1] += valid * fr;
      if (i == 0) sw0 = base0;
    }
    if (tid == 0) sb = (int)rp[0];
  }
  __syncthreads();

  // ---- fold the 2x2 output average: U[oi] = 0.5*(W[oi] + W[oi+1]) ----
  if (tid < 112) {
    int oi = tid / 16, tt = tid % 16;
    Uh[oi][tt] = 0.5f * (Wh[oi][tt] + Wh[oi + 1][tt]);
  } else if (tid < 224) {
    int q = tid - 112;
    int oj = q / 16, tt = q % 16;
    Uw[oj][tt] = 0.5f * (Ww[oj][tt] + Ww[oj + 1][tt]);
  }
  __syncthreads();

  const int wave = tid >> 5;
  const int lane = tid & 31;
  const int half = lane >> 4;   // selects K (+2) / M (+8) per WMMA f32 layouts
  const int l15 = lane & 15;
  const int h0 = sh0, w0 = sw0;
  const int colClamped = imin(w0 + l15, Wc - 1);
  float (*m1)[PITCH] = M1[wave];

  // Scalarize the per-wave channel base: sb and wave are wave-uniform, so the
  // global base pointer lives in SGPRs; with the loop fully unrolled, every
  // tile load/store is saddr + scale_offset voffset + 24-bit immediate.
  const int sb_s = __builtin_amdgcn_readfirstlane(sb);
  const int wave_s = __builtin_amdgcn_readfirstlane(wave);
  const float* fbase = feat + ((size_t)sb_s * Cc + cbase + wave_s) * HW;
  float* obase = out + ((size_t)r * Cc + cbase + wave_s) * 49;

  // Loop-invariant per-lane tile offsets (elements) for the 4 K-slices.
  // Mask is a semantic no-op (offset <= 9999) but proves the value small and
  // non-negative so every load selects the scale_offset saddr form.
  int toff0[4], toff1[4];
#pragma unroll
  for (int s = 0; s < 4; ++s) {
    const int kk = 4 * s + 2 * half;
    toff0[s] = (imin(h0 + kk, Hc - 1) * Wc + colClamped) & 16383;
    toff1[s] = (imin(h0 + kk + 1, Hc - 1) * Wc + colClamped) & 16383;
  }

  // Loop-invariant WMMA weight operands (hoisted LDS reads).
  v2f amh[4], bmw[4];
#pragma unroll
  for (int s = 0; s < 4; ++s) {
    const int kk = 4 * s + 2 * half;
    amh[s] = v2f{Uh[l15][kk], Uh[l15][kk + 1]};   // A (stage 1): lane = oi row
    bmw[s] = v2f{Uw[l15][kk], Uw[l15][kk + 1]};   // B (stage 2): B[k,n] = Uw[n][k]
  }

  // ---- software-pipelined channel walk ----
  // Preload channel 0's 16x16 tile in B-operand layout:
  // slice s, VGPR0 <- row 4s+2*half, VGPR1 <- row 4s+2*half+1, lane = tw.
  v2f bcur[4];
#pragma unroll
  for (int s = 0; s < 4; ++s)
    bcur[s] = v2f{fbase[toff0[s]], fbase[toff1[s]]};

#pragma unroll
  for (int k = 0; k < CH_PER_WAVE; ++k) {
    // Issue next channel's tile loads before any compute / DS fence so they
    // overlap the two WMMA stages and the LDS relayout of this channel.
    v2f bnxt[4];
    if (k + 1 < CH_PER_WAVE) {
      const float* fcn = fbase + (size_t)((k + 1) * NWAVES) * HW;
#pragma unroll
      for (int s = 0; s < 4; ++s)
        bnxt[s] = v2f{fcn[toff0[s]], fcn[toff1[s]]};
    }

    // ---- stage 1: M1 = Uh(16x16) x F(16x16), K = th, N = tw ----
    v8f acc = {};
#pragma unroll
    for (int s = 0; s < 4; ++s)
      acc = __builtin_amdgcn_wmma_f32_16x16x4_f32(
          false, amh[s], false, bcur[s], (short)0, acc, false, false);

    // ---- cross-lane relayout D -> A via per-wave LDS tile ----
#pragma unroll
    for (int g = 0; g < 8; ++g) m1[g + 8 * half][l15] = acc[g];
    lds_order_fence();

    // ---- stage 2: OUT = M1(16x16) x Uw^T(16x16), K = tw, N = oj ----
    v8f acc2 = {};
#pragma unroll
    for (int s = 0; s < 4; ++s) {
      const int kk = 4 * s + 2 * half;
      v2f am = v2f{m1[l15][kk], m1[l15][kk + 1]};       // A: lane = oi row
      acc2 = __builtin_amdgcn_wmma_f32_16x16x4_f32(
          false, am, false, bmw[s], (short)0, acc2, false, false);
    }
    lds_order_fence();  // keep next iteration's m1 stores behind these reads

    // ---- direct store from D layout: lane n<7 of VGPR g holds out[g][n] ----
    // obase is wave-uniform (SGPR pair); g*7 and the channel stride fold into
    // the 24-bit immediate, lane is the scale_offset voffset. The 7 stores
    // cover one contiguous 196B run -> L2 write-combines to full lines.
    if (lane < 7) {
#pragma unroll
      for (int g = 0; g < 7; ++g)
        obase[k * (NWAVES * 49) + g * 7 + lane] = acc2[g];
    }

#pragma unroll
    for (int s = 0; s < 4; ++s) bcur[s] = bnxt[s];
  }
}

extern "C" void kernel_launch(void* const* d_in, const int* in_sizes, int n_in,
                              void* d_out, int out_size, void* d_ws, size_t ws_size,
                              hipStream_t stream) {
  const float* feat = (const float*)d_in[0];
  const float* rois = (const float*)d_in[1];
  float* out = (float*)d_out;
  const int R = in_sizes[1] / 5;  // 2000
  dim3 grid(Cc / CH_PER_BLOCK, R);
  RoIAlignAvg_wmma_kernel<<<grid, 256, 0, stream>>>(feat, rois, out);
  (void)n_in; (void)out_size; (void)d_ws; (void)ws_size;
}